// XSimGCL_26199300505700
// MI455X (gfx1250) — compile-verified
//
#include <hip/hip_runtime.h>

#define NUM_USERS 100000
#define NUM_ITEMS 150000
#define N_NODES   (NUM_USERS + NUM_ITEMS)   // 250000
#define DIM       64
#define N_LAYERS  3
#define N_EDGES   2000000
#define EPS_F     0.2f
#define ROW_ELEMS (N_NODES * DIM)           // 16,000,000

typedef float v2f __attribute__((ext_vector_type(2)));
typedef float v4f __attribute__((ext_vector_type(4)));
typedef float v8f __attribute__((ext_vector_type(8)));

// ---------------------------------------------------------------------------
// WMMA row-norm helper.
// A wave owns 16 rows of 64 f32. Lane L holds row (L&15); chunk c holds
// columns 4c + koff, 4c + koff + 1 where koff = 2*(L>>4). This is exactly the
// V_WMMA_F32_16X16X4_F32 A-operand layout (VGPR0: K=k0 / k0+2, VGPR1: K=k0+1
// / k0+3). For this shape the B layout is the identical register mapping with
// N in place of M, so wmma(av, av) accumulates D = X * X^T; diag(D) = row
// squared-norms. Diagonal (m,m) lives at: m<8 -> VGPR m, lane m; m>=8 ->
// VGPR m-8, lane m+16. So lane p's useful value is d[p&7], broadcast from
// srcLane = (m<8) ? m : m+16.
// ---------------------------------------------------------------------------
__device__ __forceinline__ float sel8(const v8f d, int i) {
  float x01 = (i & 1) ? d[1] : d[0];
  float x23 = (i & 1) ? d[3] : d[2];
  float x45 = (i & 1) ? d[5] : d[4];
  float x67 = (i & 1) ? d[7] : d[6];
  float a   = (i & 2) ? x23 : x01;
  float b   = (i & 2) ? x67 : x45;
  return (i & 4) ? b : a;
}

__device__ __forceinline__ float wmma_rownormsq(const v2f av[16], int lane) {
  v8f d = {};
#pragma unroll
  for (int c = 0; c < 16; ++c) {
    // D = A * A^T + D   (16x16x4 f32 WMMA, same regs for A and B)
    d = __builtin_amdgcn_wmma_f32_16x16x4_f32(false, av[c], false, av[c],
                                              (short)0, d, false, false);
  }
  float sel = sel8(d, lane & 7);
  int   m   = lane & 15;
  int   src = (m < 8) ? m : (m + 16);
  return __shfl(sel, src, 32);
}

// ---------------------------------------------------------------------------
// Kernel 1: build all_emb = concat(user_emb, item_emb); seed acc; zero next.
// ---------------------------------------------------------------------------
__global__ void xsimgcl_init(const float* __restrict__ ue,
                             const float* __restrict__ ie,
                             float* __restrict__ cur,
                             float* __restrict__ next,
                             float* __restrict__ acc) {
  int i = blockIdx.x * blockDim.x + threadIdx.x;
  if (i >= ROW_ELEMS) return;
  float v = (i < NUM_USERS * DIM) ? ue[i] : ie[i - NUM_USERS * DIM];
  cur[i]  = v;
  acc[i]  = v;
  next[i] = 0.0f;
}

// ---------------------------------------------------------------------------
// Kernel 2: edge scatter. 16 threads per edge, float4 per thread.
// Gather hits L2-resident emb_cur (64MB << 192MB L2); atomics land in L2.
// Edge lists are stream-once -> non-temporal.
// ---------------------------------------------------------------------------
__global__ void xsimgcl_scatter(const float* __restrict__ cur,
                                float* __restrict__ next,
                                const float* __restrict__ vals,
                                const int* __restrict__ src,
                                const int* __restrict__ dst) {
  int gid = blockIdx.x * blockDim.x + threadIdx.x;
  int e   = gid >> 4;
  if (e >= N_EDGES) return;
  int c   = (gid & 15) * 4;

  int   s = __builtin_nontemporal_load(src + e);
  int   d = __builtin_nontemporal_load(dst + e);
  float v = __builtin_nontemporal_load(vals + e);

  v4f x = *(const v4f*)(cur + (size_t)s * DIM + c);
  float* out = next + (size_t)d * DIM + c;
  atomicAdd(out + 0, v * x.x);
  atomicAdd(out + 1, v * x.y);
  atomicAdd(out + 2, v * x.z);
  atomicAdd(out + 3, v * x.w);
}

// ---------------------------------------------------------------------------
// Kernel 3: e = next[r]; e += sign(e) * l2norm(noise_k[r]) * EPS;
//           cur[r] = e; acc[r] += e; (k==0) cl[r] = e; next[r] = 0.
// One wave per 16 rows; noise norms via WMMA Gram diagonal.
// ---------------------------------------------------------------------------
__global__ void xsimgcl_noise_accum(const float* __restrict__ noise_k,
                                    float* __restrict__ next,
                                    float* __restrict__ cur,
                                    float* __restrict__ acc,
                                    float* __restrict__ cl /* may be null */) {
  int lane = threadIdx.x & 31;
  int wid  = (blockIdx.x * blockDim.x + threadIdx.x) >> 5;
  int r0   = wid * 16;
  if (r0 >= N_NODES) return;               // wave-uniform: EXEC all-1s for WMMA

  int r    = r0 + (lane & 15);
  int koff = (lane >> 4) * 2;
  const float* nrow = noise_k + (size_t)r * DIM;

  v2f av[16];
#pragma unroll
  for (int c = 0; c < 16; ++c)
    av[c] = __builtin_nontemporal_load((const v2f*)(nrow + c * 4 + koff));

  float nsq = wmma_rownormsq(av, lane);
  float f   = EPS_F * rsqrtf(fmaxf(nsq, 1e-24f));

  float* erow = next + (size_t)r * DIM;
  float* crow = cur  + (size_t)r * DIM;
  float* arow = acc  + (size_t)r * DIM;
  float* lrow = cl ? (cl + (size_t)r * DIM) : nullptr;

#pragma unroll
  for (int c = 0; c < 16; ++c) {
    int off = c * 4 + koff;
    v2f e = *(const v2f*)(erow + off);
    v2f n = av[c];
    float sx = (e.x != 0.0f) ? copysignf(1.0f, e.x) : 0.0f;
    float sy = (e.y != 0.0f) ? copysignf(1.0f, e.y) : 0.0f;
    e.x += sx * n.x * f;
    e.y += sy * n.y * f;

    *(v2f*)(crow + off) = e;               // becomes input of next layer
    v2f a = *(const v2f*)(arow + off);
    a.x += e.x; a.y += e.y;
    *(v2f*)(arow + off) = a;               // acc += all_emb
    if (lrow) *(v2f*)(lrow + off) = e;     // cl snapshot (layer 0)
    v2f z = {0.0f, 0.0f};
    *(v2f*)(erow + off) = z;               // re-zero next for following layer
  }
}

// ---------------------------------------------------------------------------
// Kernel 4: buf[r] = l2norm(buf[r] * scale). WMMA row norms again.
// ---------------------------------------------------------------------------
__global__ void xsimgcl_normalize(float* __restrict__ buf, float scale) {
  int lane = threadIdx.x & 31;
  int wid  = (blockIdx.x * blockDim.x + threadIdx.x) >> 5;
  int r0   = wid * 16;
  if (r0 >= N_NODES) return;

  int r    = r0 + (lane & 15);
  int koff = (lane >> 4) * 2;
  float* row = buf + (size_t)r * DIM;

  v2f av[16];
#pragma unroll
  for (int c = 0; c < 16; ++c) {
    v2f x = *(const v2f*)(row + c * 4 + koff);
    x.x *= scale; x.y *= scale;
    av[c] = x;
  }

  float nsq = wmma_rownormsq(av, lane);
  float rn  = rsqrtf(fmaxf(nsq, 1e-24f));

#pragma unroll
  for (int c = 0; c < 16; ++c) {
    v2f x = av[c];
    x.x *= rn; x.y *= rn;
    *(v2f*)(row + c * 4 + koff) = x;
  }
}

// ---------------------------------------------------------------------------
// Launch. d_in: user_emb, item_emb, vals, noise, src, dst.
// d_out: [users|items] (light_out, 16M f32) then [users_cl|items_cl] (16M).
// d_ws : emb_cur (64MB) @ 0, emb_next (64MB) @ 64MB  (128MB total).
// ---------------------------------------------------------------------------
extern "C" void kernel_launch(void* const* d_in, const int* in_sizes, int n_in,
                              void* d_out, int out_size, void* d_ws, size_t ws_size,
                              hipStream_t stream) {
  const float* user_emb = (const float*)d_in[0];
  const float* item_emb = (const float*)d_in[1];
  const float* vals     = (const float*)d_in[2];
  const float* noise    = (const float*)d_in[3];   // (3, N_NODES, 64)
  const int*   src      = (const int*)d_in[4];
  const int*   dst      = (const int*)d_in[5];

  float* out0 = (float*)d_out;                 // light_out (pre-normalize: acc)
  float* out1 = out0 + (size_t)ROW_ELEMS;      // cl_emb

  float* cur  = (float*)d_ws;
  float* next = cur + (size_t)ROW_ELEMS;

  const int BLK = 256;
  const int initBlocks  = (ROW_ELEMS + BLK - 1) / BLK;            // 62500
  const int scatBlocks  = (N_EDGES * 16) / BLK;                   // 125000
  const int waveBlocks  = ((N_NODES + 15) / 16 * 32 + BLK - 1) / BLK; // 1954

  xsimgcl_init<<<initBlocks, BLK, 0, stream>>>(user_emb, item_emb, cur, next, out0);

  for (int k = 0; k < N_LAYERS; ++k) {
    xsimgcl_scatter<<<scatBlocks, BLK, 0, stream>>>(cur, next, vals, src, dst);
    xsimgcl_noise_accum<<<waveBlocks, BLK, 0, stream>>>(
        noise + (size_t)k * ROW_ELEMS, next, cur, out0,
        (k == 0) ? out1 : nullptr);
  }

  xsimgcl_normalize<<<waveBlocks, BLK, 0, stream>>>(out0, 0.25f);  // acc/4 then l2norm
  xsimgcl_normalize<<<waveBlocks, BLK, 0, stream>>>(out1, 1.0f);   // l2norm(cl)
}